// ChessGNN_10015863734961
// MI455X (gfx1250) — compile-verified
//
#include <hip/hip_runtime.h>
#include <hip/hip_bf16.h>

typedef __attribute__((ext_vector_type(2))) float v2f;
typedef __attribute__((ext_vector_type(8))) float v8f;

// ---------------------------------------------------------------------------
// zero workspace region
// ---------------------------------------------------------------------------
__global__ void gcn_zero_kernel(float* __restrict__ p, long long n) {
    long long i = (long long)blockIdx.x * blockDim.x + threadIdx.x;
    long long stride = (long long)gridDim.x * blockDim.x;
    for (; i < n; i += stride) p[i] = 0.0f;
}

// ---------------------------------------------------------------------------
// degree accumulation: deg[dst] += 1 for every edge (self loop added later)
// ---------------------------------------------------------------------------
__global__ void gcn_degree_kernel(const int* __restrict__ dst,
                                  float* __restrict__ deg, int E) {
    int e = blockIdx.x * blockDim.x + threadIdx.x;
    if (e < E) atomicAdd(&deg[dst[e]], 1.0f);
}

// deg -> dinv = rsqrt(deg + 1)   (+1 = self loop)
__global__ void gcn_dinv_kernel(float* __restrict__ d, int n) {
    int i = blockIdx.x * blockDim.x + threadIdx.x;
    if (i < n) d[i] = rsqrtf(d[i] + 1.0f);
}

// ---------------------------------------------------------------------------
// Dense per-node GEMM with V_WMMA_F32_16X16X4_F32 (full fp32 precision).
// Each wave computes a 16(node) x 32(out) tile:  out = (X @ W) * dinv[node]
//   A 16x4 fp32 layout: lanes 0-15 -> (M=lane, K=4s..4s+1) in a.x/a.y
//                       lanes 16-31 -> (M=lane-16, K=4s+2..4s+3)
//   B 4x16 fp32 layout mirrors A; C/D: VGPR r, lane l -> M=r+8*(l>=16), N=l%16
// ---------------------------------------------------------------------------
template <int K>
__global__ void gcn_gemm_scale_kernel(const float* __restrict__ X,   // [n,K]
                                      const float* __restrict__ W,   // [K,32]
                                      const float* __restrict__ dinv,
                                      float* __restrict__ out,       // [n,32]
                                      int n) {
    const int lane = threadIdx.x & 31;
    const int wave = threadIdx.x >> 5;
    const int hi   = lane >> 4;   // half-wave select
    const int lo   = lane & 15;
    const int base = (blockIdx.x * (blockDim.x >> 5) + wave) * 16;
    if (base >= n) return;        // wave-uniform branch: EXEC stays all-ones

    v8f c0 = {};
    v8f c1 = {};
    const float* xrow = X + (long long)(base + lo) * K;

#pragma unroll
    for (int s = 0; s < K / 4; ++s) {
        const int k0 = s * 4 + 2 * hi;
        v2f a;
        a.x = xrow[k0];
        a.y = xrow[k0 + 1];
        v2f b0, b1v;
        b0.x  = W[(k0) * 32 + lo];
        b0.y  = W[(k0 + 1) * 32 + lo];
        b1v.x = W[(k0) * 32 + 16 + lo];
        b1v.y = W[(k0 + 1) * 32 + 16 + lo];
        // D = A*B + C, fp32 throughout: v_wmma_f32_16x16x4_f32
        c0 = __builtin_amdgcn_wmma_f32_16x16x4_f32(
                 false, a, false, b0, (short)0, c0, false, false);
        c1 = __builtin_amdgcn_wmma_f32_16x16x4_f32(
                 false, a, false, b1v, (short)0, c1, false, false);
    }

#pragma unroll
    for (int r = 0; r < 8; ++r) {
        const int m = base + r + 8 * hi;
        const float sc = dinv[m];
        out[(long long)m * 32 + lo]      = c0[r] * sc;
        out[(long long)m * 32 + 16 + lo] = c1[r] * sc;
    }
}

// ---------------------------------------------------------------------------
// Edge scatter: acc[dst] += hs[src]  (32 floats per edge; 8 threads/edge,
// float4 each -> contiguous within an edge, coalesced across 8-thread groups)
// ---------------------------------------------------------------------------
__global__ void gcn_scatter_kernel(const int* __restrict__ src,
                                   const int* __restrict__ dst,
                                   const float* __restrict__ hs,
                                   float* __restrict__ acc, int E) {
    long long tid = (long long)blockIdx.x * blockDim.x + threadIdx.x;
    int e = (int)(tid >> 3);
    int q = (int)(tid & 7);
    if (e >= E) return;
    const int s = src[e];
    const int d = dst[e];
    const float4 v = *(const float4*)(hs + (long long)s * 32 + q * 4);
    float* p = acc + (long long)d * 32 + q * 4;
    atomicAdd(p + 0, v.x);
    atomicAdd(p + 1, v.y);
    atomicAdd(p + 2, v.z);
    atomicAdd(p + 3, v.w);
}

// ---------------------------------------------------------------------------
// finalize layer 1: h1f = relu(dinv*(acc + hs_self) + b); acc re-zeroed for L2
// ---------------------------------------------------------------------------
__global__ void gcn_finalize_relu_kernel(const float* __restrict__ dinv,
                                         float* __restrict__ acc,
                                         const float* __restrict__ hs,
                                         const float* __restrict__ bias,
                                         float* __restrict__ out, int n) {
    int tid = blockIdx.x * blockDim.x + threadIdx.x;
    if (tid >= n * 32) return;
    const int i = tid >> 5;
    const int c = tid & 31;
    const float v = dinv[i] * (acc[tid] + hs[tid]) + bias[c];
    out[tid] = fmaxf(v, 0.0f);
    acc[tid] = 0.0f;   // ready for layer 2 aggregation
}

// ---------------------------------------------------------------------------
// finalize layer 2 fused with global max pool (relu output >= 0 so bitwise
// int atomicMax == float max; pool pre-zeroed)
// ---------------------------------------------------------------------------
__global__ void gcn_finalize_pool_kernel(const float* __restrict__ dinv,
                                         const float* __restrict__ acc,
                                         const float* __restrict__ hs,
                                         const float* __restrict__ bias,
                                         float* __restrict__ pool, int n) {
    int tid = blockIdx.x * blockDim.x + threadIdx.x;
    if (tid >= n * 32) return;
    const int i = tid >> 5;
    const int c = tid & 31;
    const float v = fmaxf(dinv[i] * (acc[tid] + hs[tid]) + bias[c], 0.0f);
    atomicMax((int*)&pool[c], __float_as_int(v));
}

// ---------------------------------------------------------------------------
// tiny head: logits = pool @ fcW + fcb; log_softmax -> d_out[0..4]
// ---------------------------------------------------------------------------
__global__ void gcn_head_kernel(const float* __restrict__ pool,
                                const float* __restrict__ fcW,
                                const float* __restrict__ fcb,
                                float* __restrict__ out) {
    if (threadIdx.x != 0) return;
    float l[5];
#pragma unroll
    for (int j = 0; j < 5; ++j) {
        float s = fcb[j];
#pragma unroll
        for (int c = 0; c < 32; ++c) s += pool[c] * fcW[c * 5 + j];
        l[j] = s;
    }
    float m = l[0];
#pragma unroll
    for (int j = 1; j < 5; ++j) m = fmaxf(m, l[j]);
    float se = 0.0f;
#pragma unroll
    for (int j = 0; j < 5; ++j) se += expf(l[j] - m);
    const float lse = logf(se);
#pragma unroll
    for (int j = 0; j < 5; ++j) out[j] = l[j] - m - lse;
}

// ---------------------------------------------------------------------------
// launch
// ---------------------------------------------------------------------------
extern "C" void kernel_launch(void* const* d_in, const int* in_sizes, int n_in,
                              void* d_out, int out_size, void* d_ws, size_t ws_size,
                              hipStream_t stream) {
    const float* x   = (const float*)d_in[0];   // [n,8]
    const int*   ei  = (const int*)d_in[1];     // [2,E]
    const float* W1  = (const float*)d_in[2];   // [8,32]
    const float* b1  = (const float*)d_in[3];   // [32]
    const float* W2  = (const float*)d_in[4];   // [32,32]
    const float* b2  = (const float*)d_in[5];   // [32]
    const float* fcW = (const float*)d_in[6];   // [32,5]
    const float* fcb = (const float*)d_in[7];   // [5]

    const int n = in_sizes[0] / 8;   // 100000 nodes
    const int E = in_sizes[1] / 2;   // 1600000 edges
    const int* src = ei;
    const int* dst = ei + E;

    // workspace layout (floats): [dinv n][acc n*32][pool 32][hs n*32][h1f n*32]
    float* ws   = (float*)d_ws;
    float* dinv = ws;
    float* acc  = dinv + n;
    float* pool = acc + (long long)n * 32;
    float* hs   = pool + 32;
    float* h1f  = hs + (long long)n * 32;

    const long long zero_n = (long long)n * 33 + 32;  // dinv + acc + pool
    gcn_zero_kernel<<<1024, 256, 0, stream>>>(ws, zero_n);

    gcn_degree_kernel<<<(E + 255) / 256, 256, 0, stream>>>(dst, dinv, E);
    gcn_dinv_kernel<<<(n + 255) / 256, 256, 0, stream>>>(dinv, n);

    const int gemm_grid  = (n + 127) / 128;                 // 8 waves x 16 nodes
    const int scat_grid  = (int)(((long long)E * 8 + 255) / 256);
    const int elem_grid  = (n * 32 + 255) / 256;

    // ---- layer 1 ----
    gcn_gemm_scale_kernel<8><<<gemm_grid, 256, 0, stream>>>(x, W1, dinv, hs, n);
    gcn_scatter_kernel<<<scat_grid, 256, 0, stream>>>(src, dst, hs, acc, E);
    gcn_finalize_relu_kernel<<<elem_grid, 256, 0, stream>>>(dinv, acc, hs, b1, h1f, n);

    // ---- layer 2 ----
    gcn_gemm_scale_kernel<32><<<gemm_grid, 256, 0, stream>>>(h1f, W2, dinv, hs, n);
    gcn_scatter_kernel<<<scat_grid, 256, 0, stream>>>(src, dst, hs, acc, E);
    gcn_finalize_pool_kernel<<<elem_grid, 256, 0, stream>>>(dinv, acc, hs, b2, pool, n);

    // ---- head ----
    gcn_head_kernel<<<1, 32, 0, stream>>>(pool, fcW, fcb, (float*)d_out);
}